// MobiusConv_52261162057779
// MI455X (gfx1250) — compile-verified
//
#include <hip/hip_runtime.h>
#include <hip/hip_bf16.h>

// ---------------------------------------------------------------------------
// MobiusConv (linearBasisConvLT forward) for gfx1250 / MI455X
//
// out[b,o,q,m,t,s] = sum_{i,u,v} x*E1*E2*F[q,m,u,v]*W[i,u,v,o]
// Factorization:
//   G[(i,u,v)] = c0 * tu[u] * pv[v]
//   c0  = x * exp(-sig*lnA + (T+sig)*lnT) * cis(s_q*(lnT-lnA) + mInd*(Psi-Phi))
//   tu  = {1, conj(ct), ct},  ct = cis(lnT)    (pTau index: u'=0,1,-1)
//   pv  = {cp, 1, conj(cp)},  cp = cis(Psi)    (pPsi index: v'=-1,0,1)
// Final contraction per (q,m): real GEMM  M=2048 x K=144 x N=32
//   (K = re/im-interleaved (i,u,v);  N = [o_re(0..15) | o_im(16..31)])
//   B[2k  ][o_re] =  Re(F*W)   B[2k+1][o_re] = -Im(F*W)
//   B[2k  ][o_im] =  Im(F*W)   B[2k+1][o_im] =  Re(F*W)
// computed with V_WMMA_F32_16X16X4_F32 (fp32 in / fp32 acc).
// ---------------------------------------------------------------------------

typedef __attribute__((ext_vector_type(2))) float v2f;
typedef __attribute__((ext_vector_type(8))) float v8f;

#define BATCHN 2
#define ICN 8
#define OCN 16
#define TDIM 32                 // 2B
#define QN 20
#define M2N 6
#define NCELL (BATCHN*TDIM*TDIM*ICN)   // 16384 (pixel, i) cells
#define SPN (BATCHN*TDIM*TDIM)         // 2048 spatial rows
#define KDIM 144                       // (IC * 9 uv) * 2 (re/im)
#define NDIM 32                        // oC re + oC im
#define QMN (QN*M2N)                   // 120
#define ROWS 64                        // spatial rows per workgroup
#define LDS_STRIDE 148                 // padded row stride (conflict-free)

struct c2 { float x, y; };
__device__ __forceinline__ c2 cmul(c2 a, c2 b) {
    return { a.x*b.x - a.y*b.y, a.x*b.y + a.y*b.x };
}

// ---------------------------------------------------------------------------
// Phase 1a: per (pixel,i) cell precompute.
// ws0[cell*8 + {0..7}] = {mag, lnT-lnA, Psi-Phi, cos(lnT), sin(lnT),
//                         cos(Psi), sin(Psi), 0}
// ---------------------------------------------------------------------------
__global__ void precomp_cells(const float* __restrict__ x,
                              const float* __restrict__ lnA,
                              const float* __restrict__ Phi,
                              const float* __restrict__ lnT,
                              const float* __restrict__ Psi,
                              const float* __restrict__ pAlpha,  // complex64
                              const float* __restrict__ pTau,    // complex64
                              float* __restrict__ ws0) {
    int c = blockIdx.x * blockDim.x + threadIdx.x;
    if (c >= NCELL) return;
    float sig = -pAlpha[0];   // Re(pAlpha[0,0]) = -sigma
    float tps =  pTau[0];     // Re(pTau[0,0,0,0]) = T + sigma
    float la = lnA[c], lt = lnT[c], ph = Phi[c], ps = Psi[c];
    float mag = x[c] * expf(-sig * la + tps * lt);
    float stc, ctc, spc, cpc;
    sincosf(lt, &stc, &ctc);
    sincosf(ps, &spc, &cpc);
    float* w = ws0 + (size_t)c * 8;
    w[0] = mag; w[1] = lt - la; w[2] = ps - ph;
    w[3] = ctc; w[4] = stc; w[5] = cpc; w[6] = spc; w[7] = 0.f;
}

// ---------------------------------------------------------------------------
// Phase 1b: pack B = F[q,m,u,v] * W[i,u,v,o] into ws1[qm][n][k]
// (transposed [n][k] so main-kernel B-operand loads are contiguous float2).
// W from collectWeights: j=4 -> (coeffs[i,0,o], 0);
//   j<4 -> jj=3-j, j>4 -> jj=j-5; W = coeffs[i,1+2jj,o] + i*coeffs[i,2+2jj,o]
// ---------------------------------------------------------------------------
__global__ void build_B(const float* __restrict__ coeffs,  // (8,9,16) f32
                        const float* __restrict__ F,       // (20,6,3,3) c64
                        float* __restrict__ ws1) {
    int qm = blockIdx.x;  // 0..119
    for (int l = threadIdx.x; l < NDIM * KDIM; l += blockDim.x) {
        int n = l / KDIM, k = l % KDIM;
        int p = k & 1;              // 0 = re-part row of G, 1 = im-part row
        int j = (k >> 1) % 9;       // uv index (u*3+v)
        int i = (k >> 1) / 9;       // input channel
        int o = n & 15, cpart = n >> 4;
        int jj = (j == 4) ? -1 : (j < 4 ? 3 - j : j - 5);
        float wr, wi;
        if (jj < 0) { wr = coeffs[(i*9 + 0)*16 + o]; wi = 0.f; }
        else {
            wr = coeffs[(i*9 + 1 + 2*jj)*16 + o];
            wi = coeffs[(i*9 + 2 + 2*jj)*16 + o];
        }
        int fidx = (qm * 9 + j) * 2;
        float fr = F[fidx], fi = F[fidx + 1];
        float FWr = fr*wr - fi*wi;
        float FWi = fr*wi + fi*wr;
        float val = (cpart == 0) ? (p == 0 ?  FWr : -FWi)
                                 : (p == 0 ?  FWi :  FWr);
        ws1[((size_t)qm * NDIM + n) * KDIM + k] = val;
    }
}

// ---------------------------------------------------------------------------
// Main kernel: grid = (SPN/ROWS = 32 chunks, QM = 120), block = 128 (4 waves)
// ---------------------------------------------------------------------------
__global__ __launch_bounds__(128) void mobius_main(
        const float* __restrict__ ws0, const float* __restrict__ ws1,
        const float* __restrict__ pAlpha, const float* __restrict__ pPhi,
        float* __restrict__ out) {
    __shared__ float Gs[ROWS][LDS_STRIDE];   // A panel (64 x 144, padded)
    __shared__ float Bt[NDIM][LDS_STRIDE];   // B transposed (32 x 144, padded)

    int qm = blockIdx.y;
    int q = qm / M2N, m = qm - q * M2N;
    int chunk = blockIdx.x;
    int tid = threadIdx.x;

    float sq   = -pAlpha[(q * M2N) * 2 + 1];  // s_q = -Im(pAlpha[q,0])
    float mind = -pPhi[m * 2 + 1];            // mInd = -Im(pPhi[0,m])

    // ---- stage B panel into LDS (float4 copies; 144 % 4 == 0 keeps rows) ----
    const float* Bg = ws1 + (size_t)qm * NDIM * KDIM;
    for (int l4 = tid; l4 < (NDIM * KDIM) / 4; l4 += 128) {
        int l = l4 * 4;
        int n = l / KDIM, k = l % KDIM;
        float4 v = *(const float4*)(Bg + l);
        *(float4*)&Bt[n][k] = v;
    }

    // ---- build G panel: 64 rows x 8 channels = 512 tasks, 4 per thread -----
    #pragma unroll
    for (int it = 0; it < (ROWS * ICN) / 128; ++it) {
        int task = it * 128 + tid;
        int i = task & 7, row = task >> 3;
        int sp = chunk * ROWS + row;
        const float* cell = ws0 + ((size_t)sp * ICN + i) * 8;
        float mag = cell[0], dlt = cell[1], dpp = cell[2];
        float ctc = cell[3], cts = cell[4], cpc = cell[5], cps = cell[6];
        float th = sq * dlt + mind * dpp;
        float sn, cs;
        sincosf(th, &sn, &cs);
        c2 c0 = { mag * cs, mag * sn };
        c2 tu[3] = { {1.f, 0.f}, {ctc, -cts}, {ctc, cts} };   // u' = 0, 1, -1
        c2 pv[3] = { {cpc, cps}, {1.f, 0.f}, {cpc, -cps} };   // v' = -1, 0, 1
        int kb = i * 18;
        #pragma unroll
        for (int u = 0; u < 3; ++u) {
            c2 cu = cmul(c0, tu[u]);
            #pragma unroll
            for (int v = 0; v < 3; ++v) {
                c2 g = cmul(cu, pv[v]);
                *(float2*)&Gs[row][kb + (u * 3 + v) * 2] = make_float2(g.x, g.y);
            }
        }
    }
    __syncthreads();

    // ---- WMMA: each wave owns one 16-row M-tile x both 16-col N-tiles ------
    int wave = tid >> 5, lane = tid & 31;
    int l15 = lane & 15;
    bool hi = lane >= 16;
    int rowBase = wave * 16;
    int rA = rowBase + l15;
    int koff = hi ? 2 : 0;       // A/B layout: lanes 16-31 carry K+2 / K+3

    v8f accR = {};
    v8f accI = {};
    for (int kk = 0; kk < KDIM / 4; ++kk) {
        int k0 = kk * 4 + koff;
        v2f a  = *(const v2f*)&Gs[rA][k0];
        v2f bR = *(const v2f*)&Bt[l15][k0];
        v2f bI = *(const v2f*)&Bt[16 + l15][k0];
        accR = __builtin_amdgcn_wmma_f32_16x16x4_f32(
                   false, a, false, bR, (short)0, accR, false, false);
        accI = __builtin_amdgcn_wmma_f32_16x16x4_f32(
                   false, a, false, bI, (short)0, accI, false, false);
    }

    // ---- write out: lane L holds N=L&15; VGPR r -> M = r (+8 for hi half) --
    int o = l15;
    int rAdd = hi ? 8 : 0;
    #pragma unroll
    for (int r = 0; r < 8; ++r) {
        int sp = chunk * ROWS + rowBase + r + rAdd;
        int b = sp >> 10;            // / (32*32)
        int ts = sp & 1023;
        size_t idx = ((((size_t)(b * OCN + o) * QN + q) * M2N + m) * 1024 + ts) * 2;
        *(float2*)(out + idx) = make_float2(accR[r], accI[r]);
    }
}

// ---------------------------------------------------------------------------
extern "C" void kernel_launch(void* const* d_in, const int* in_sizes, int n_in,
                              void* d_out, int out_size, void* d_ws, size_t ws_size,
                              hipStream_t stream) {
    const float* x      = (const float*)d_in[0];
    const float* lnA    = (const float*)d_in[1];
    const float* Phi    = (const float*)d_in[2];
    const float* lnT    = (const float*)d_in[3];
    const float* Psi    = (const float*)d_in[4];
    const float* coeffs = (const float*)d_in[5];
    const float* pAlpha = (const float*)d_in[6];   // complex64 -> float pairs
    const float* pPhi   = (const float*)d_in[7];
    const float* pTau   = (const float*)d_in[8];
    // d_in[9] = pPsi (folded analytically), d_in[10] = F
    const float* F      = (const float*)d_in[10];
    float* out = (float*)d_out;

    float* ws0 = (float*)d_ws;                 // 16384 * 8 f32 = 512 KB
    float* ws1 = ws0 + (size_t)NCELL * 8;      // 120 * 32 * 144 f32 = 2.21 MB

    precomp_cells<<<NCELL / 256, 256, 0, stream>>>(x, lnA, Phi, lnT, Psi,
                                                   pAlpha, pTau, ws0);
    build_B<<<QMN, 256, 0, stream>>>(coeffs, F, ws1);
    dim3 grid(SPN / ROWS, QMN);
    mobius_main<<<grid, 128, 0, stream>>>(ws0, ws1, pAlpha, pPhi, out);
}